// SAFM8_41266045780172
// MI455X (gfx1250) — compile-verified
//
#include <hip/hip_runtime.h>
#include <math.h>

// ---------------------------------------------------------------------------
// SAFM8: multi-scale depthwise refinement + per-pixel CxC softmax mixing +
// 1x1 conv (WMMA fp32) + exact GELU * x.   B=4, C=32, H=W=128.
// ---------------------------------------------------------------------------
#define BB 4
#define CC 32
#define HH 128
#define WW 128
#define HWHW (HH * WW)

typedef __attribute__((ext_vector_type(2))) float v2f;
typedef __attribute__((ext_vector_type(8))) float v8f;

// ---------------- 2x2 max pool (stride 2) ----------------------------------
__global__ void safm_pool2(const float* __restrict__ in, float* __restrict__ out,
                           int Hin, int Win) {
    int Ho = Hin >> 1, Wo = Win >> 1;
    int total = BB * CC * Ho * Wo;
    int idx = blockIdx.x * blockDim.x + threadIdx.x;
    if (idx >= total) return;
    int wo = idx % Wo;
    int t = idx / Wo;
    int ho = t % Ho;
    int bc = t / Ho;
    const float* p = in + ((long)bc * Hin + ho * 2) * Win + wo * 2;
    float m = fmaxf(fmaxf(p[0], p[1]), fmaxf(p[Win], p[Win + 1]));
    out[idx] = m;
}

// ---------------- depthwise 3x3, pad 1 --------------------------------------
__global__ void safm_dwconv(const float* __restrict__ in, const float* __restrict__ w,
                            const float* __restrict__ bias, float* __restrict__ out,
                            int Hs, int Ws) {
    int total = BB * CC * Hs * Ws;
    int idx = blockIdx.x * blockDim.x + threadIdx.x;
    if (idx >= total) return;
    int x = idx % Ws;
    int t = idx / Ws;
    int y = t % Hs;
    int bc = t / Hs;
    int c = bc % CC;
    const float* plane = in + (long)bc * Hs * Ws;
    const float* wc = w + c * 9;
    float acc = bias[c];
#pragma unroll
    for (int ky = 0; ky < 3; ky++) {
        int yy = y + ky - 1;
        if (yy < 0 || yy >= Hs) continue;
#pragma unroll
        for (int kx = 0; kx < 3; kx++) {
            int xx = x + kx - 1;
            if (xx < 0 || xx >= Ws) continue;
            acc = fmaf(plane[yy * Ws + xx], wc[ky * 3 + kx], acc);
        }
    }
    out[idx] = acc;
}

// ---------------- per-pixel CxC softmax mixing (wave32 = 32 channels) -------
// qkv[n,i] = sum_j exp(v_i k_j - m_j) * q_j / (sum_i' exp(v_i' k_j - m_j))
__global__ void safm_attn(const float* __restrict__ s0, const float* __restrict__ s1,
                          const float* __restrict__ s2, float* __restrict__ qkv) {
    int gtid = blockIdx.x * blockDim.x + threadIdx.x;
    int n = gtid >> 5;            // pixel
    int lane = gtid & 31;         // channel
    if (n >= BB * HWHW) return;
    int b = n / HWHW;
    int hw = n % HWHW;
    int h = hw / WW, w = hw % WW;

    float q = s0[((long)b * CC + lane) * HWHW + hw];
    int hw1 = (h >> 1) * (WW >> 1) + (w >> 1);
    float k = s1[((long)b * CC + lane) * (HWHW >> 2) + hw1];
    int hw2 = (h >> 2) * (WW >> 2) + (w >> 2);
    float v = s2[((long)b * CC + lane) * (HWHW >> 4) + hw2];

    // column max of v_i*k_j is (k_j>=0 ? vmax : vmin)*k_j ; softmax shift-invariant
    float vmax = v, vmin = v;
#pragma unroll
    for (int off = 16; off >= 1; off >>= 1) {
        vmax = fmaxf(vmax, __shfl_xor(vmax, off, 32));
        vmin = fminf(vmin, __shfl_xor(vmin, off, 32));
    }

    float acc = 0.f;
#pragma unroll 4
    for (int j = 0; j < 32; j++) {
        float kj = __shfl(k, j, 32);
        float qj = __shfl(q, j, 32);
        float m = (kj >= 0.f) ? vmax * kj : vmin * kj;
        float e = __expf(fmaf(v, kj, -m));
        float S = e;
#pragma unroll
        for (int off = 16; off >= 1; off >>= 1) S += __shfl_xor(S, off, 32);
        acc = fmaf(e, qj / S, acc);
    }
    qkv[(long)n * CC + lane] = acc;
}

// ---------------- 1x1 conv via WMMA f32 16x16x4 + GELU(exact) * x -----------
// out[n,o] = sum_c qkv[n,c] * wa[o,c] + ba[o]; one wave = 16 pixels x 32 outs.
__global__ void safm_gemm_epi(const float* __restrict__ qkv, const float* __restrict__ wa,
                              const float* __restrict__ ba, const float* __restrict__ x,
                              float* __restrict__ out) {
    __shared__ float lds[8][32 * 16];   // per-wave transpose buffer [o][pix]
    int waveInBlk = threadIdx.x >> 5;
    int lane = threadIdx.x & 31;
    int tile = blockIdx.x * (blockDim.x >> 5) + waveInBlk;   // 16-pixel tile
    const int nTiles = BB * HWHW / 16;                       // 4096 (grid exact)
    if (tile >= nTiles) return;

    int pixBase = tile * 16;
    int mrow = lane & 15;              // M (pixel) for A / N (out-ch) for B
    int khalf = (lane >> 4) * 2;       // ISA A layout: lanes 16-31 hold K+2

    v8f acc0 = {};    // outputs o = 0..15
    v8f acc1 = {};    // outputs o = 16..31
#pragma unroll
    for (int k0 = 0; k0 < 32; k0 += 4) {
        v2f a;
        const float* arow = qkv + (long)(pixBase + mrow) * CC + k0 + khalf;
        a.x = arow[0];
        a.y = arow[1];
        v2f b0, b1;   // B[k][n] = wa[n][k] (wa is (O,C) row-major)
        b0.x = wa[mrow * CC + k0 + khalf];
        b0.y = wa[mrow * CC + k0 + khalf + 1];
        b1.x = wa[(16 + mrow) * CC + k0 + khalf];
        b1.y = wa[(16 + mrow) * CC + k0 + khalf + 1];
        acc0 = __builtin_amdgcn_wmma_f32_16x16x4_f32(false, a, false, b0,
                                                     (short)0, acc0, false, false);
        acc1 = __builtin_amdgcn_wmma_f32_16x16x4_f32(false, a, false, b1,
                                                     (short)0, acc1, false, false);
    }

    // C/D layout: VGPR r, lanes 0-15 -> M=r, N=lane ; lanes 16-31 -> M=r+8
    float* T = &lds[waveInBlk][0];     // T[o*16 + pix]
#pragma unroll
    for (int r = 0; r < 8; r++) {
        int pix = r + 8 * (lane >> 4);
        int o = lane & 15;
        T[o * 16 + pix] = acc0[r];
        T[(o + 16) * 16 + pix] = acc1[r];
    }
    __syncthreads();

    // Tiles of 16 pixels lie in one image row (W=128 % 16 == 0) -> coalesced.
    int b = pixBase / HWHW;
    int hw = pixBase % HWHW;
#pragma unroll
    for (int it = 0; it < 16; it++) {
        int o = it * 2 + (lane >> 4);
        int pix = lane & 15;
        float val = T[o * 16 + pix] + ba[o];
        float g = 0.5f * val * (1.0f + erff(val * 0.70710678118654752f)); // exact GELU
        long gidx = ((long)b * CC + o) * HWHW + hw + pix;
        out[gidx] = g * x[gidx];
    }
}

// ---------------------------------------------------------------------------
extern "C" void kernel_launch(void* const* d_in, const int* in_sizes, int n_in,
                              void* d_out, int out_size, void* d_ws, size_t ws_size,
                              hipStream_t stream) {
    const float* x  = (const float*)d_in[0];
    const float* w0 = (const float*)d_in[1];
    const float* b0 = (const float*)d_in[2];
    const float* w1 = (const float*)d_in[3];
    const float* b1 = (const float*)d_in[4];
    const float* w2 = (const float*)d_in[5];
    const float* b2 = (const float*)d_in[6];
    const float* wa = (const float*)d_in[7];
    const float* ba = (const float*)d_in[8];
    float* out = (float*)d_out;

    char* ws = (char*)d_ws;
    float* s0  = (float*)(ws);                               // 8 MB  (B,C,128,128)
    float* p1  = (float*)(ws + (8u << 20));                  // 2 MB  (B,C,64,64)
    float* s1s = (float*)(ws + (10u << 20));                 // 2 MB
    float* p2  = (float*)(ws + (12u << 20));                 // 0.5MB (B,C,32,32)
    float* s2s = (float*)(ws + (12u << 20) + (1u << 19));    // 0.5MB
    float* qkv = (float*)(ws + (13u << 20));                 // 8 MB  (N,C)

    // level 0 conv
    safm_dwconv<<<8192, 256, 0, stream>>>(x, w0, b0, s0, HH, WW);
    // level 1: pool2 -> conv
    safm_pool2<<<2048, 256, 0, stream>>>(x, p1, HH, WW);
    safm_dwconv<<<2048, 256, 0, stream>>>(p1, w1, b1, s1s, 64, 64);
    // level 2: pool2(pool2) -> conv
    safm_pool2<<<512, 256, 0, stream>>>(p1, p2, 64, 64);
    safm_dwconv<<<512, 256, 0, stream>>>(p2, w2, b2, s2s, 32, 32);
    // per-pixel CxC softmax mixing (one wave per pixel)
    safm_attn<<<8192, 256, 0, stream>>>(s0, s1s, s2s, qkv);
    // 1x1 conv via WMMA + bias + exact GELU * x  (4096 tiles, 8 waves/block)
    safm_gemm_epi<<<512, 256, 0, stream>>>(qkv, wa, ba, x, out);
}